// rgcn_link_predict_40578851013127
// MI455X (gfx1250) — compile-verified
//
#include <hip/hip_runtime.h>

// ---------------------------------------------------------------------------
// RGCN (2-layer, basis-decomposed) forward for MI455X / gfx1250.
//
//   W1  = comp1 @ basis1            ([40x30]@[30x1M])      -> ws (160 MB)
//   cnt = per-(dst,rel) edge count  -> cntInv = 1/max(cnt,1)
//   h   = relu( scatter(W1[rel,src]*cntInv) + root1 + bias1 )
//   W2  = comp2 @ basis2            ([40x30]@[30x10000])
//   Hx  = h @ W2[r]  for all r      (40 GEMMs)             -> ws (reuse W1)
//   x   = scatter(Hx[rel,src]*cntInv) + h@root2 + bias2    -> d_out[0:1M)
//   emb = emb_rel[edge_type] (row 0 zeroed)                -> d_out[1M:33M)
//
// All GEMMs use V_WMMA_F32_16X16X4_F32 with *unconditional* loads:
//  - B of h@W2 staged in LDS zero-padded to 112 cols (conflict-free ds reads)
//  - A of comp@basis staged transposed+zero-padded in LDS; K-tail reads valid
//    B rows against zeroed A (0*finite == 0), so no predication, no OOB.
// ---------------------------------------------------------------------------

#define N_NODES 10000
#define N_REL   40
#define N_BASES 30
#define HIDDEN  100
#define N_EDGES 320000
#define LDSB_COLS 112   // HIDDEN padded to multiple of 16

typedef __attribute__((ext_vector_type(2))) float v2f;
typedef __attribute__((ext_vector_type(8))) float v8f;

__device__ __forceinline__ v8f wmma_f32(v2f a, v2f b, v8f c) {
    return __builtin_amdgcn_wmma_f32_16x16x4_f32(false, a, false, b,
                                                 (short)0, c, false, false);
}

// ---------------------------------------------------------------------------
// Out[z] = A @ B[z] (+bias):  A [M x 100] (M % 16 == 0), B[z] [100 x 100].
// grid: (ceil(mtiles/8), ztiles). One wave per 16-row stripe; B tile in LDS.
// ---------------------------------------------------------------------------
__global__ __launch_bounds__(256) void gemm_nodes(
    const float* __restrict__ A, int mtiles,
    const float* __restrict__ B, long long bzStride,
    float* __restrict__ Out, long long ozStride,
    const float* __restrict__ bias)
{
    __shared__ float ldsB[HIDDEN * LDSB_COLS];
    const int z = blockIdx.y;
    const float* Bz = B + (long long)z * bzStride;

    for (int i = threadIdx.x; i < HIDDEN * LDSB_COLS; i += blockDim.x) {
        const int row = i / LDSB_COLS, col = i % LDSB_COLS;
        ldsB[i] = (col < HIDDEN) ? Bz[row * HIDDEN + col] : 0.0f;
    }
    __syncthreads();

    const int mt = blockIdx.x * (blockDim.x >> 5) + (threadIdx.x >> 5);
    if (mt >= mtiles) return;                       // wave-uniform
    const int  lane   = threadIdx.x & 31;
    const int  lanelo = lane & 15;
    const bool hi     = lane >= 16;
    const int  kOff   = hi ? 2 : 0;
    const int  m      = mt * 16 + lanelo;

    // Preload this lane's A fragment: pairs (4k+kOff, 4k+kOff+1), k=0..24.
    float2 areg[25];
    const float* Arow = A + (long long)m * HIDDEN + kOff;
    #pragma unroll
    for (int kk = 0; kk < 25; ++kk)
        areg[kk] = *(const float2*)(Arow + kk * 4);

    float* Oz = Out + (long long)z * ozStride;
    #pragma unroll 1
    for (int nt = 0; nt < 7; ++nt) {
        const int col = nt * 16 + lanelo;           // <= 111, LDS-padded
        v8f acc = {};
        #pragma unroll
        for (int kk = 0; kk < 25; ++kk) {
            const int kb = kk * 4 + kOff;
            v2f a; a.x = areg[kk].x; a.y = areg[kk].y;
            v2f b;
            b.x = ldsB[kb * LDSB_COLS + col];
            b.y = ldsB[(kb + 1) * LDSB_COLS + col];
            acc = wmma_f32(a, b, acc);
        }
        if (col < HIDDEN) {
            const float bv = bias ? bias[col] : 0.0f;
            const int rbase = mt * 16 + (hi ? 8 : 0);
            #pragma unroll
            for (int j = 0; j < 8; ++j)
                Oz[(long long)(rbase + j) * HIDDEN + col] = acc[j] + bv;
        }
    }
}

// ---------------------------------------------------------------------------
// Out = A @ B:  A [Ma x 30] tiny (comp), B [30 x N] streamed, N % 16 == 0.
// A staged transposed + zero-padded [32 x 48] in LDS. K padded 30 -> 32; the
// padded k rows have A==0, so B re-reads valid rows 28/29 there (0*x == 0).
// ---------------------------------------------------------------------------
__global__ __launch_bounds__(256) void gemm_comp(
    const float* __restrict__ A, int Ma,
    const float* __restrict__ B, long long N,
    float* __restrict__ Out, int ntiles)
{
    __shared__ float ldsAT[32 * 48];
    for (int i = threadIdx.x; i < 32 * 48; i += blockDim.x) {
        const int k = i / 48, m = i % 48;
        ldsAT[i] = (k < N_BASES && m < Ma) ? A[m * N_BASES + k] : 0.0f;
    }
    __syncthreads();

    const int mtiles = 3;
    const int w = blockIdx.x * (blockDim.x >> 5) + (threadIdx.x >> 5);
    if (w >= mtiles * ntiles) return;               // wave-uniform
    const int nt = w / mtiles;
    const int mt = w % mtiles;
    const int  lane   = threadIdx.x & 31;
    const int  lanelo = lane & 15;
    const bool hi     = lane >= 16;
    const int  kOff   = hi ? 2 : 0;
    const int  m      = mt * 16 + lanelo;
    const long long col = (long long)nt * 16 + lanelo;

    v8f acc = {};
    #pragma unroll
    for (int kk = 0; kk < 8; ++kk) {
        const int ka = kk * 4 + kOff;               // kk==7,hi: ka=30/31 (A==0)
        v2f a, b;
        a.x = ldsAT[ka * 48 + m];
        a.y = ldsAT[(ka + 1) * 48 + m];
        const long long r0 = (kk < 7) ? ka : 28;    // valid rows for padded k
        b.x = B[r0 * N + col];
        b.y = B[(r0 + 1) * N + col];
        acc = wmma_f32(a, b, acc);
    }
    const int rbase = mt * 16 + (hi ? 8 : 0);
    #pragma unroll
    for (int j = 0; j < 8; ++j) {
        const int r = rbase + j;
        if (r < Ma) Out[(long long)r * N + col] = acc[j];
    }
}

// --------------------------- small helper kernels ---------------------------

__global__ void kzero(float* __restrict__ p, int n) {
    int i = blockIdx.x * blockDim.x + threadIdx.x;
    if (i < n) p[i] = 0.0f;
}

__global__ void kcount(const int* __restrict__ dst, const int* __restrict__ etype,
                       float* __restrict__ cnt, int nEdges) {
    int e = blockIdx.x * blockDim.x + threadIdx.x;
    if (e < nEdges) unsafeAtomicAdd(&cnt[dst[e] * N_REL + etype[e]], 1.0f);
}

__global__ void kinv(float* __restrict__ cnt, int n) {
    int i = blockIdx.x * blockDim.x + threadIdx.x;
    if (i < n) cnt[i] = 1.0f / fmaxf(cnt[i], 1.0f);
}

__global__ void kinith(const float* __restrict__ root1, const float* __restrict__ bias1,
                       float* __restrict__ hbuf, int n) {
    int i = blockIdx.x * blockDim.x + threadIdx.x;
    if (i < n) hbuf[i] = root1[i] + bias1[i % HIDDEN];
}

__global__ void krelu(float* __restrict__ p, int n) {
    int i = blockIdx.x * blockDim.x + threadIdx.x;
    if (i < n) p[i] = fmaxf(p[i], 0.0f);
}

// target[dst] += msg[rel*N*H + src*H + :] * cntInv[dst,rel]; float4 per thread.
__global__ void kscatter(const float* __restrict__ msg, const float* __restrict__ cntInv,
                         const int* __restrict__ src, const int* __restrict__ dst,
                         const int* __restrict__ etype, float* __restrict__ target,
                         int nEdges) {
    const int QUADS = HIDDEN / 4;                   // 25 float4 per edge
    long long i = (long long)blockIdx.x * blockDim.x + threadIdx.x;
    if (i >= (long long)nEdges * QUADS) return;
    int e = (int)(i / QUADS);
    int q = (int)(i % QUADS);
    int r = etype[e], s = src[e], d = dst[e];
    float w = cntInv[d * N_REL + r];
    float4 v = ((const float4*)(msg + (long long)r * (N_NODES * HIDDEN)
                                    + (long long)s * HIDDEN))[q];
    float* t = target + (long long)d * HIDDEN + q * 4;
    unsafeAtomicAdd(t + 0, v.x * w);
    unsafeAtomicAdd(t + 1, v.y * w);
    unsafeAtomicAdd(t + 2, v.z * w);
    unsafeAtomicAdd(t + 3, v.w * w);
}

__global__ void kemb(const float* __restrict__ emb, const int* __restrict__ etype,
                     float* __restrict__ out, int nEdges) {
    const int QUADS = HIDDEN / 4;
    long long i = (long long)blockIdx.x * blockDim.x + threadIdx.x;
    if (i >= (long long)nEdges * QUADS) return;
    int e = (int)(i / QUADS);
    int q = (int)(i % QUADS);
    int r = etype[e];
    float4 v = make_float4(0.f, 0.f, 0.f, 0.f);
    if (r != 0) v = ((const float4*)emb)[r * QUADS + q];
    ((float4*)out)[i] = v;
}

// ---------------------------------------------------------------------------

extern "C" void kernel_launch(void* const* d_in, const int* in_sizes, int n_in,
                              void* d_out, int out_size, void* d_ws, size_t ws_size,
                              hipStream_t stream) {
    const float* comp1  = (const float*)d_in[0];   // [40,30]
    const float* basis1 = (const float*)d_in[1];   // [30,10000,100] == [30, 1e6]
    const float* root1  = (const float*)d_in[2];   // [10000,100]
    const float* bias1  = (const float*)d_in[3];   // [100]
    const float* comp2  = (const float*)d_in[4];   // [40,30]
    const float* basis2 = (const float*)d_in[5];   // [30,100,100] == [30, 10000]
    const float* root2  = (const float*)d_in[6];   // [100,100]
    const float* bias2  = (const float*)d_in[7];   // [100]
    const float* embrel = (const float*)d_in[8];   // [41,100]
    const int*   eidx   = (const int*)d_in[9];     // [2,320000]
    const int*   etype  = (const int*)d_in[10];    // [320000]

    const int* src = eidx;
    const int* dst = eidx + N_EDGES;

    float* xout   = (float*)d_out;                     // [10000,100]
    float* embout = (float*)d_out + N_NODES * HIDDEN;  // [320000,100]

    const long long NH   = (long long)N_NODES * HIDDEN;          // 1,000,000
    const long long W1SZ = (long long)N_REL * NH;                // 40,000,000
    float* W1buf  = (float*)d_ws;                                // W1, then Hx
    float* hbuf   = W1buf + W1SZ;
    float* W2buf  = hbuf + NH;
    float* cntbuf = W2buf + (long long)N_REL * HIDDEN * HIDDEN;
    const size_t needBytes = (size_t)(W1SZ + NH + 400000 + 400000) * sizeof(float);
    if (ws_size < needBytes) return;

    const int TPB = 256;
    const int NSEG = N_NODES * N_REL;
    const long long EQ = (long long)N_EDGES * (HIDDEN / 4);
    const int MTILES = N_NODES / 16;                             // 625
    const int MBLK = (MTILES + 7) / 8;                           // 79

    // per-(dst,rel) counts -> reciprocals
    kzero <<<(NSEG + TPB - 1) / TPB, TPB, 0, stream>>>(cntbuf, NSEG);
    kcount<<<(N_EDGES + TPB - 1) / TPB, TPB, 0, stream>>>(dst, etype, cntbuf, N_EDGES);
    kinv  <<<(NSEG + TPB - 1) / TPB, TPB, 0, stream>>>(cntbuf, NSEG);

    // W1 = comp1 @ basis1   [40x30] @ [30 x 1e6]
    {
        const int nt = (int)(NH / 16);                           // 62500
        const int waves = 3 * nt;
        gemm_comp<<<(waves + 7) / 8, TPB, 0, stream>>>(
            comp1, N_REL, basis1, NH, W1buf, nt);
    }
    // W2 = comp2 @ basis2   [40x30] @ [30 x 10000]
    {
        const int nt = (HIDDEN * HIDDEN) / 16;                   // 625
        const int waves = 3 * nt;
        gemm_comp<<<(waves + 7) / 8, TPB, 0, stream>>>(
            comp2, N_REL, basis2, HIDDEN * HIDDEN, W2buf, nt);
    }

    // h = relu( root1 + bias1 + scatter(W1[rel,src] * cntInv) )
    kinith<<<(int)((NH + TPB - 1) / TPB), TPB, 0, stream>>>(root1, bias1, hbuf, (int)NH);
    kscatter<<<(int)((EQ + TPB - 1) / TPB), TPB, 0, stream>>>(
        W1buf, cntbuf, src, dst, etype, hbuf, N_EDGES);
    krelu<<<(int)((NH + TPB - 1) / TPB), TPB, 0, stream>>>(hbuf, (int)NH);

    // Hx[r] = h @ W2[r]  (40 GEMMs) -> reuse W1buf
    gemm_nodes<<<dim3(MBLK, N_REL), TPB, 0, stream>>>(
        hbuf, MTILES, W2buf, (long long)HIDDEN * HIDDEN, W1buf, NH, nullptr);

    // x = h @ root2 + bias2  (straight into d_out)
    gemm_nodes<<<dim3(MBLK, 1), TPB, 0, stream>>>(
        hbuf, MTILES, root2, 0, xout, 0, bias2);

    // x += scatter(Hx[rel,src] * cntInv)
    kscatter<<<(int)((EQ + TPB - 1) / TPB), TPB, 0, stream>>>(
        W1buf, cntbuf, src, dst, etype, xout, N_EDGES);

    // relation embedding lookup (padding row 0)
    kemb<<<(int)((EQ + TPB - 1) / TPB), TPB, 0, stream>>>(embrel, etype, embout, N_EDGES);
}